// Trainer_61993557950978
// MI455X (gfx1250) — compile-verified
//
#include <hip/hip_runtime.h>
#include <math.h>

typedef __attribute__((ext_vector_type(2))) float v2f;
typedef __attribute__((ext_vector_type(8))) float v8f;

#define DIM   4096
#define NSEG  (DIM / 16)          // 256 segments per channel
#define NTILE (NSEG / 16)         // 16 WMMA tiles (16 segments each) per channel
#define NTHR  1024                // 32 waves, one WGP

// Fully fused kernel: prefix sums over k (tensor-pipe scan), per-batch gather,
// deterministic block reduction, final scalar math. One launch, one workgroup.
__global__ __launch_bounds__(NTHR)
void Trainer_61993557950978_kernel(const float* __restrict__ betas,
                                   const float* __restrict__ lambdas,
                                   const float* __restrict__ gammas,
                                   const float* __restrict__ hx,
                                   const float* __restrict__ ixy,
                                   float* __restrict__ out, int B) {
    // Layout: sv0 | sv1 | sv2 (3 x 4096 prefix channels) | st (3 x 256 seg totals)
    __shared__ float smem[3 * DIM + 3 * NSEG];
    float* sv = smem;                 // sv + c*DIM
    float* st = smem + 3 * DIM;       // st + c*NSEG

    const int tid  = threadIdx.x;
    const int wid  = tid >> 5;
    const int lane = tid & 31;
    const int lo   = lane & 15;       // column / M index
    const int hi   = lane >> 4;       // half-wave selector (K pair / M half)

    // ---- Phase 1: per-k channel values -------------------------------------
    // ch0 = gamma_k, ch1 = gamma_k*log(lambda_k), ch2 = gamma_k*log(1-lambda_k)
    for (int k = tid; k < DIM; k += NTHR) {
        float g = gammas[k];
        float l = lambdas[k];
        sv[0 * DIM + k] = g;
        sv[1 * DIM + k] = g * __logf(l);
        sv[2 * DIM + k] = g * __logf(1.0f - l);
    }
    __syncthreads();

    // ---- Phase 2: in-segment inclusive scans on the matrix pipe ------------
    // S = L16 x X, L16 lower-triangular ones, via 4 chained wmma_f32_16x16x4_f32.
    // Tile t covers 16 segments (columns n=0..15), 16 rows each, in place.
    for (int t = wid; t < 3 * NTILE; t += (NTHR / 32)) {
        const int c  = t >> 4;        // channel
        const int tt = t & 15;        // tile within channel
        float* base = sv + c * DIM + tt * 256;   // X[r][n] = base[n*16 + r]
        v8f acc = {};
#pragma unroll
        for (int j = 0; j < 4; ++j) {
            const int k0 = 4 * j + 2 * hi;       // this lane's K pair
            v2f a, b;
            a.x = (k0     <= lo) ? 1.0f : 0.0f;  // L[m=lo][k0]
            a.y = (k0 + 1 <= lo) ? 1.0f : 0.0f;  // L[m=lo][k0+1]
            b.x = base[lo * 16 + k0];            // X[k0  ][n=lo]
            b.y = base[lo * 16 + k0 + 1];        // X[k0+1][n=lo]
            acc = __builtin_amdgcn_wmma_f32_16x16x4_f32(
                false, a, false, b, (short)0, acc, false, false);
        }
        // D layout: vgpr v -> row (hi*8 + v), column lo
#pragma unroll
        for (int v = 0; v < 8; ++v)
            base[lo * 16 + hi * 8 + v] = acc[v];
    }
    __syncthreads();

    // ---- Phase 3: scan of segment totals (Hillis-Steele, 256/channel) ------
    {
        const bool act = tid < 3 * NSEG;
        const int c = tid >> 8, s = tid & 255;
        if (act) st[c * NSEG + s] = sv[c * DIM + s * 16 + 15];
        __syncthreads();
        for (int off = 1; off < NSEG; off <<= 1) {
            float v = 0.0f;
            if (act) {
                v = st[c * NSEG + s];
                if (s >= off) v += st[c * NSEG + s - off];
            }
            __syncthreads();
            if (act) st[c * NSEG + s] = v;
            __syncthreads();
        }
    }

    // ---- Phase 4: add exclusive segment offsets ----------------------------
    for (int i = tid; i < 3 * DIM; i += NTHR) {
        const int c = i >> 12, k = i & (DIM - 1), seg = k >> 4;
        if (seg > 0) sv[c * DIM + k] += st[c * NSEG + seg - 1];
    }
    __syncthreads();

    // ---- Phase 5: per-batch gather + local accumulation --------------------
    float s_ixt = 0.0f, s_nI = 0.0f, s_LG = 0.0f, s_LC = 0.0f;
    for (int b = tid; b < B; b += NTHR) {
        const float beta = betas[b];
        int kb = (int)floorf(beta - 1.0f);
        kb = max(0, min(kb, DIM));
        if (kb >= 1) {
            const float P0 = sv[0 * DIM + kb - 1];   // G(kb)
            const float P1 = sv[1 * DIM + kb - 1];   // sum gamma*log(lambda)
            const float P2 = sv[2 * DIM + kb - 1];   // sum gamma*log(1-lambda)
            const float lam_max = 1.0f - 1.0f / ((float)kb + 1.0f);
            const float f = __logf(lam_max) - __logf(1.0f - lam_max);
            s_ixt += P0 * f + (P2 - P1);
            s_nI  += P0;
            s_LG  += P1;
            s_LC  += P2;
        }
    }
    __syncthreads();   // done reading sv; reuse its LDS for the reduction

    // ---- Phase 6: deterministic block reduction of 4 scalars ---------------
    smem[tid]            = s_ixt;
    smem[NTHR + tid]     = s_nI;
    smem[2 * NTHR + tid] = s_LG;
    smem[3 * NTHR + tid] = s_LC;
    __syncthreads();
    for (int s = NTHR / 2; s > 0; s >>= 1) {
        if (tid < s) {
            smem[tid]            += smem[tid + s];
            smem[NTHR + tid]     += smem[NTHR + tid + s];
            smem[2 * NTHR + tid] += smem[2 * NTHR + tid + s];
            smem[3 * NTHR + tid] += smem[3 * NTHR + tid + s];
        }
        __syncthreads();
    }

    // ---- Phase 7: final scalar math ----------------------------------------
    if (tid == 0) {
        const float ixt = smem[0];
        const float nI  = smem[NTHR];
        const float SL  = smem[2 * NTHR];
        const float SC  = smem[3 * NTHR];

        const float gm_term  = __expf(SL / nI);
        const float gm_comp  = __expf(SC / nI);
        const float exp_term = __expf(2.0f * ixt / nI);
        const float chechik  = -nI * 0.5f * __logf(gm_comp + exp_term * gm_term);
        const float ity = ixt + chechik;

        const float rhs = 1.0f - ity / ixy[0];
        float lhs1 = 1.0f - ixt / hx[0];
        if (lhs1 < 0.0f) lhs1 = fabsf(lhs1) * 20.0f;
        const float Cc = 0.42062202f;               // exp(-0.866)
        const float lhs = Cc * __powf(lhs1, 1.92f);

        out[0] = rhs;
        out[1] = lhs;
    }
}

extern "C" void kernel_launch(void* const* d_in, const int* in_sizes, int n_in,
                              void* d_out, int out_size, void* d_ws, size_t ws_size,
                              hipStream_t stream) {
    const float* betas   = (const float*)d_in[0];
    const float* lambdas = (const float*)d_in[1];
    const float* gammas  = (const float*)d_in[2];
    const float* hx      = (const float*)d_in[3];
    const float* ixy     = (const float*)d_in[4];
    float* out = (float*)d_out;
    const int B = in_sizes[0];

    Trainer_61993557950978_kernel<<<1, NTHR, 0, stream>>>(
        betas, lambdas, gammas, hx, ixy, out, B);
}